// AttentionGraphModel_24129126269419
// MI455X (gfx1250) — compile-verified
//
#include <hip/hip_runtime.h>
#include <hip/hip_bf16.h>
#include <math.h>

#define NNODES 50000
#define NEDGES 800000
#define DIM    128
#define NLAYERS 3

typedef float v2f __attribute__((ext_vector_type(2)));
typedef float v8f __attribute__((ext_vector_type(8)));

// ---------------------------------------------------------------------------
// Kernel 1: H = leaky_relu(X @ W + b, 0.2) via V_WMMA_F32_16X16X4_F32.
// Block = 256 threads (8 wave32). Block owns rows [16*bx, 16*bx+16), each
// wave owns one 16-column tile. A-strip (16x128) staged in LDS with padded
// stride 140 (bank-conflict-free lane-strided fragment reads, 16B aligned).
// ---------------------------------------------------------------------------
#define AS_STRIDE 140

__global__ __launch_bounds__(256) void gat_linear_wmma(
    const float* __restrict__ X, const float* __restrict__ W,
    const float* __restrict__ bias, float* __restrict__ Hout)
{
    __shared__ float As[16][AS_STRIDE];

    const int tid  = threadIdx.x;
    const int wave = tid >> 5;
    const int lane = tid & 31;
    const int mbase = blockIdx.x * 16;

    // Cooperative load of the 16x128 A strip: 2048 floats = 512 float4,
    // 256 threads x 2 float4 each. 8 consecutive floats stay in one row.
    {
        int g = tid * 8;                 // 0..2040, row-aligned (128 % 8 == 0)
        int row = g >> 7;                // g / 128
        int col = g & 127;
        const float4* src = (const float4*)(X + (size_t)(mbase + row) * DIM + col);
        float4* dst = (float4*)(&As[row][col]);
        dst[0] = src[0];
        dst[1] = src[1];
    }
    __syncthreads();

    const int halfsel = lane >> 4;       // 0: K pair {0,1}; 1: K pair {2,3}
    const int l15     = lane & 15;       // A: row M;  B/D: column N (within tile)
    const int colbase = wave * 16;

    v8f acc = {0.f, 0.f, 0.f, 0.f, 0.f, 0.f, 0.f, 0.f};

    #pragma unroll 8
    for (int t = 0; t < 32; ++t) {
        const int k = 4 * t + 2 * halfsel;
        // A fragment (ISA 7.12.2, 32-bit A 16x4): lanes 0-15 hold K=0/1,
        // lanes 16-31 hold K=2/3; VGPR index selects within the pair.
        v2f a;
        a.x = As[l15][k];
        a.y = As[l15][k + 1];
        // B fragment (4x16, row striped across lanes): lanes 0-15 row K=k,
        // lanes 16-31 row K=k (with k already offset by halfsel); VGPR1 = K+1.
        const float* wp = W + (size_t)k * DIM + colbase + l15;
        v2f b;
        b.x = wp[0];
        b.y = wp[DIM];
        acc = __builtin_amdgcn_wmma_f32_16x16x4_f32(
            /*neg_a=*/false, a, /*neg_b=*/false, b,
            /*c_mod=*/(short)0, acc, /*reuse_a=*/false, /*reuse_b=*/false);
    }

    // Epilogue: C/D layout -> VGPR r: lanes 0-15 row M=r, lanes 16-31 row M=r+8.
    const int col  = colbase + l15;
    const float bj = bias[col];
    const int row0 = mbase + 8 * halfsel;
    #pragma unroll
    for (int r = 0; r < 8; ++r) {
        float v = acc[r] + bj;
        v = (v > 0.f) ? v : 0.2f * v;    // leaky_relu(0.2)
        Hout[(size_t)(row0 + r) * DIM + col] = v;
    }
}

// ---------------------------------------------------------------------------
// Kernel 2: per-node attention scores  sl[i] = h[i].aw_left, sr[i] = h[i].aw_right
// One wave32 per node: 32 lanes x float4 = 128 elements, shfl reduction.
// ---------------------------------------------------------------------------
__global__ __launch_bounds__(128) void gat_scores(
    const float* __restrict__ H, const float* __restrict__ aw,  // aw: 2*DIM floats
    float* __restrict__ sl, float* __restrict__ sr)
{
    const int wave = threadIdx.x >> 5;
    const int lane = threadIdx.x & 31;
    const int node = blockIdx.x * 4 + wave;
    if (node >= NNODES) return;

    const float4 h4 = ((const float4*)(H + (size_t)node * DIM))[lane];
    const float4 wl = ((const float4*)(aw))[lane];
    const float4 wr = ((const float4*)(aw + DIM))[lane];
    float pl = h4.x * wl.x + h4.y * wl.y + h4.z * wl.z + h4.w * wl.w;
    float pr = h4.x * wr.x + h4.y * wr.y + h4.z * wr.z + h4.w * wr.w;
    #pragma unroll
    for (int off = 16; off > 0; off >>= 1) {
        pl += __shfl_down(pl, off, 32);
        pr += __shfl_down(pr, off, 32);
    }
    if (lane == 0) { sl[node] = pl; sr[node] = pr; }
}

// ---------------------------------------------------------------------------
// Order-preserving uint encoding of float (for segment-max via atomicMax).
// enc(-inf) = 0x007FFFFF > 0, so initializing to 0 is a safe "-infinity".
// ---------------------------------------------------------------------------
__device__ __forceinline__ unsigned enc_f(float f) {
    unsigned u = __float_as_uint(f);
    return (u & 0x80000000u) ? ~u : (u | 0x80000000u);
}
__device__ __forceinline__ float dec_f(unsigned u) {
    return __uint_as_float((u & 0x80000000u) ? (u & 0x7FFFFFFFu) : ~u);
}

// Kernel 3: zero the per-layer accumulators (Q = h2, segment max, denom)
__global__ void gat_init(float* __restrict__ Q, unsigned* __restrict__ menc,
                         float* __restrict__ denom)
{
    const size_t i = (size_t)blockIdx.x * blockDim.x + threadIdx.x;
    if (i < (size_t)NNODES * DIM) Q[i] = 0.f;
    if (i < NNODES) { menc[i] = 0u; denom[i] = 0.f; }
}

// Kernel 4: e_ij = sl[src] + sr[dst] + b ; segment-max over src via atomicMax
__global__ void gat_edge_logits(
    const int* __restrict__ src, const int* __restrict__ dst,
    const float* __restrict__ sl, const float* __restrict__ sr,
    const float* __restrict__ ab, float* __restrict__ e,
    unsigned* __restrict__ menc)
{
    const int i = blockIdx.x * blockDim.x + threadIdx.x;
    if (i >= NEDGES) return;
    const int s = src[i];
    const float v = sl[s] + sr[dst[i]] + ab[0];
    e[i] = v;
    atomicMax(menc + s, enc_f(v));
}

// Kernel 5: ex = exp(e - m[src]) ; denom[src] += ex
__global__ void gat_edge_exp(
    const int* __restrict__ src, const unsigned* __restrict__ menc,
    float* __restrict__ e, float* __restrict__ denom)
{
    const int i = blockIdx.x * blockDim.x + threadIdx.x;
    if (i >= NEDGES) return;
    const int s = src[i];
    const float ex = __expf(e[i] - dec_f(menc[s]));
    e[i] = ex;
    unsafeAtomicAdd(denom + s, ex);   // -> global_atomic_add_f32
}

// Kernel 6: h2[src] += h[dst] * (ex / denom[src]).  One wave32 per edge:
// 32 lanes x float4 cover the 128-dim row; 4 hw f32 atomics per lane into
// the L2-resident 25.6MB accumulator.
__global__ __launch_bounds__(256) void gat_aggregate(
    const int* __restrict__ src, const int* __restrict__ dst,
    const float* __restrict__ ex, const float* __restrict__ denom,
    const float* __restrict__ H, float* __restrict__ Q)
{
    const int wave = threadIdx.x >> 5;
    const int lane = threadIdx.x & 31;
    const long long eidx = (long long)blockIdx.x * 8 + wave;
    if (eidx >= NEDGES) return;
    const int s = src[eidx];
    const int d = dst[eidx];
    const float alpha = ex[eidx] / denom[s];
    const float4 h4 = ((const float4*)(H + (size_t)d * DIM))[lane];
    float* qp = Q + (size_t)s * DIM + 4 * lane;
    unsafeAtomicAdd(qp + 0, h4.x * alpha);
    unsafeAtomicAdd(qp + 1, h4.y * alpha);
    unsafeAtomicAdd(qp + 2, h4.z * alpha);
    unsafeAtomicAdd(qp + 3, h4.w * alpha);
}

// Kernel 7: elementwise ReLU (in-place for intermediate layers, -> d_out last)
__global__ void gat_relu(const float* __restrict__ in, float* __restrict__ out)
{
    const size_t i = (size_t)blockIdx.x * blockDim.x + threadIdx.x;
    if (i < (size_t)NNODES * DIM) out[i] = fmaxf(in[i], 0.f);
}

// ---------------------------------------------------------------------------
// Launcher
// ---------------------------------------------------------------------------
extern "C" void kernel_launch(void* const* d_in, const int* in_sizes, int n_in,
                              void* d_out, int out_size, void* d_ws, size_t ws_size,
                              hipStream_t stream)
{
    const float* x      = (const float*)d_in[0];
    const int*   esrc   = (const int*)  d_in[1];
    const int*   edst   = (const int*)  d_in[2];
    const float* lin_w  = (const float*)d_in[3];   // [3,128,128] row-major [in,out]
    const float* lin_b  = (const float*)d_in[4];   // [3,128]
    const float* attn_w = (const float*)d_in[5];   // [3,256]
    const float* attn_b = (const float*)d_in[6];   // [3]
    float* out = (float*)d_out;

    // Workspace carve-out (~55 MB)
    char* ws = (char*)d_ws;
    float*    P     = (float*)ws;    ws += sizeof(float) * (size_t)NNODES * DIM; // post-linear h
    float*    Q     = (float*)ws;    ws += sizeof(float) * (size_t)NNODES * DIM; // h2 / next input
    float*    e     = (float*)ws;    ws += sizeof(float) * (size_t)NEDGES;       // logits -> exp
    float*    sl    = (float*)ws;    ws += sizeof(float) * (size_t)NNODES;
    float*    sr    = (float*)ws;    ws += sizeof(float) * (size_t)NNODES;
    unsigned* menc  = (unsigned*)ws; ws += sizeof(unsigned) * (size_t)NNODES;
    float*    denom = (float*)ws;

    const int elemBlocks = (NNODES * DIM + 255) / 256;   // 25000
    const int edgeBlocks = (NEDGES + 255) / 256;         // 3125
    const int aggBlocks  = (NEDGES + 7) / 8;             // 100000

    const float* cur = x;
    for (int l = 0; l < NLAYERS; ++l) {
        gat_linear_wmma<<<NNODES / 16, 256, 0, stream>>>(
            cur, lin_w + (size_t)l * DIM * DIM, lin_b + (size_t)l * DIM, P);
        gat_scores<<<NNODES / 4, 128, 0, stream>>>(
            P, attn_w + (size_t)l * 2 * DIM, sl, sr);
        gat_init<<<elemBlocks, 256, 0, stream>>>(Q, menc, denom);
        gat_edge_logits<<<edgeBlocks, 256, 0, stream>>>(
            esrc, edst, sl, sr, attn_b + l, e, menc);
        gat_edge_exp<<<edgeBlocks, 256, 0, stream>>>(esrc, menc, e, denom);
        gat_aggregate<<<aggBlocks, 256, 0, stream>>>(esrc, edst, e, denom, P, Q);
        float* reluDst = (l == NLAYERS - 1) ? out : Q;
        gat_relu<<<elemBlocks, 256, 0, stream>>>(Q, reluDst);
        cur = Q;
    }
}